// KWinners_74569222193245
// MI455X (gfx1250) — compile-verified
//
#include <hip/hip_runtime.h>
#include <math.h>
#include <stdint.h>

// ---------------------------------------------------------------------------
// K-winners (top-K per sample with per-channel boost) via radix-select.
//   B=64 samples, N=32*32*512=524288 units/sample, K=52428, C=512 channels.
// Rounds: 11 bits -> 11 bits -> 10 bits of the order-preserving uint key.
// Workspace: hist[B][2048] + prefix/remain/tie[B] + boost[C]  (~527 KB).
// Histogram passes stream global data through LDS with gfx1250 async copies
// (ASYNCcnt-tracked, double-buffered: fetch it+1 overlaps binning of it).
// ---------------------------------------------------------------------------

static constexpr int      kB   = 64;
static constexpr int      kC   = 512;
static constexpr int      kN   = 32 * 32 * 512;     // 524288
static constexpr unsigned kK   = 52428u;
static constexpr int      kBins = 2048;
static constexpr int      kChunk = 1024;            // elements per block-iteration
static constexpr int      kEPB   = 16384;           // elements per hist block
static constexpr int      kHistBlocksPerSample = kN / kEPB;   // 32
static constexpr int      kIters = kEPB / kChunk;   // 16

// Monotonic float -> uint mapping (larger float => larger key).
__device__ __forceinline__ unsigned orderedKey(float f) {
  unsigned u = __float_as_uint(f);
  return (u & 0x80000000u) ? ~u : (u | 0x80000000u);
}

// 16-byte-per-lane async global->LDS copy (gfx1250, ASYNCcnt tracked).
__device__ __forceinline__ void asyncLoad16(const float* gsrc, float* ldsDst) {
#if defined(__gfx1250__)
  unsigned lds = (unsigned)(uintptr_t)ldsDst;             // low 32 bits = LDS offset
  unsigned long long gaddr = (unsigned long long)(uintptr_t)gsrc;
  asm volatile("global_load_async_to_lds_b128 %0, %1, off"
               :: "v"(lds), "v"(gaddr)
               : "memory");
#else
  ldsDst[0] = gsrc[0]; ldsDst[1] = gsrc[1];
  ldsDst[2] = gsrc[2]; ldsDst[3] = gsrc[3];
#endif
}

// Wait until at most CNT async ops are outstanding for this wave.
template <int CNT>
__device__ __forceinline__ void waitAsyncLe() {
#if defined(__gfx1250__)
  if constexpr (CNT == 0) asm volatile("s_wait_asynccnt 0x0" ::: "memory");
  else                    asm volatile("s_wait_asynccnt 0x1" ::: "memory");
#endif
}

// ---------------------------------------------------------------------------
// Init: zero histograms, reset per-sample state, precompute boost factors.
// ---------------------------------------------------------------------------
__global__ __launch_bounds__(256) void kw_init(unsigned* __restrict__ hist,
                                               unsigned* __restrict__ prefix,
                                               unsigned* __restrict__ remain,
                                               unsigned* __restrict__ tie,
                                               float* __restrict__ boost,
                                               const float* __restrict__ duty) {
  int i = blockIdx.x * blockDim.x + threadIdx.x;
  if (i < kB * kBins) hist[i] = 0u;
  if (i < kB) { prefix[i] = 0u; remain[i] = kK; tie[i] = 0u; }
  if (i < kC) {
    float target = (float)kK / (float)kN;
    boost[i] = expf((target - duty[i]) * 1.0f);
  }
}

// ---------------------------------------------------------------------------
// Histogram pass (templated on radix round). Double-buffered async staging:
// no intra-loop barriers -- each thread consumes only its own staged 16B, so
// the per-wave ASYNCcnt wait is the only ordering required.
// ---------------------------------------------------------------------------
template <int ROUND>
__global__ __launch_bounds__(256) void kw_hist(const float* __restrict__ in,
                                               const float* __restrict__ boost,
                                               unsigned* __restrict__ hist,
                                               const unsigned* __restrict__ prefix) {
  __shared__ unsigned lbin[kBins];
  __shared__ float    bst[kC];
  __shared__ float    stage[2][kChunk];

  const int b     = blockIdx.x / kHistBlocksPerSample;
  const int chunk = blockIdx.x % kHistBlocksPerSample;
  const int tid   = threadIdx.x;

  for (int i = tid; i < kBins; i += 256) lbin[i] = 0u;
  for (int i = tid; i < kC;    i += 256) bst[i]  = boost[i];

  unsigned pfx = 0u;
  if (ROUND > 0) pfx = prefix[b];

  const float* src = in + (size_t)b * kN + (size_t)chunk * kEPB;

  __syncthreads();                                  // lbin/bst ready

  // Prologue: kick off iteration 0's copy.
  asyncLoad16(src + tid * 4, &stage[0][tid * 4]);

  for (int it = 0; it < kIters; ++it) {
    const int cur = it & 1;
    if (it + 1 < kIters) {
      asyncLoad16(src + (it + 1) * kChunk + tid * 4, &stage[cur ^ 1][tid * 4]);
      waitAsyncLe<1>();                             // iter `it` landed; it+1 in flight
    } else {
      waitAsyncLe<0>();                             // drain
    }

    float4 v = *reinterpret_cast<const float4*>(&stage[cur][tid * 4]);
    float vals[4] = {v.x, v.y, v.z, v.w};
#pragma unroll
    for (int j = 0; j < 4; ++j) {
      int c = (tid * 4 + j) & (kC - 1);             // chunk base is a multiple of 1024
      unsigned key = orderedKey(vals[j] * bst[c]);
      bool ok;
      unsigned digit;
      if (ROUND == 0)      { ok = true;                 digit = key >> 21; }
      else if (ROUND == 1) { ok = ((key >> 21) == pfx); digit = (key >> 10) & 0x7FFu; }
      else                 { ok = ((key >> 10) == pfx); digit = key & 0x3FFu; }
      if (ok) atomicAdd(&lbin[digit], 1u);
    }
  }
  __syncthreads();
  for (int i = tid; i < kBins; i += 256) {
    unsigned cnt = lbin[i];
    if (cnt) atomicAdd(&hist[b * kBins + i], cnt);
  }
}

// ---------------------------------------------------------------------------
// Select: one block per sample. Finds the bin holding the K-th largest key,
// narrows prefix/remaining, and zeroes the histogram for the next round.
// ---------------------------------------------------------------------------
__global__ __launch_bounds__(256) void kw_select(unsigned* __restrict__ hist,
                                                 unsigned* __restrict__ prefix,
                                                 unsigned* __restrict__ remain,
                                                 int nbins, int shiftBits) {
  __shared__ unsigned cnt[kBins];
  const int b   = blockIdx.x;
  const int tid = threadIdx.x;
  for (int i = tid; i < nbins; i += 256) {
    cnt[i] = hist[b * kBins + i];
    hist[b * kBins + i] = 0u;              // ready for next round
  }
  __syncthreads();
  if (tid == 0) {
    unsigned R = remain[b];
    unsigned cum = 0u, greater = 0u;
    int sel = 0;
    for (int d = nbins - 1; d >= 0; --d) {
      unsigned nc = cum + cnt[d];
      if (nc >= R) { sel = d; greater = cum; break; }
      cum = nc;
    }
    prefix[b] = (prefix[b] << shiftBits) | (unsigned)sel;
    remain[b] = R - greater;               // winners still needed from this bin
  }
}

// ---------------------------------------------------------------------------
// Output: gate ORIGINAL inputs. key > thr -> keep; key == thr -> first
// `remain` ticket holders keep (exactly K ones per sample); else zero.
// ---------------------------------------------------------------------------
__global__ __launch_bounds__(256) void kw_output(const float* __restrict__ in,
                                                 const float* __restrict__ boost,
                                                 const unsigned* __restrict__ prefix,
                                                 const unsigned* __restrict__ remain,
                                                 unsigned* __restrict__ tie,
                                                 float* __restrict__ out) {
  __shared__ float bst[kC];
  const int tid = threadIdx.x;
  for (int i = tid; i < kC; i += 256) bst[i] = boost[i];
  __syncthreads();

  const int blocksPerSample = kN / kChunk;           // 512
  const int b = blockIdx.x / blocksPerSample;
  const size_t base = (size_t)blockIdx.x * kChunk + (size_t)tid * 4;

  const unsigned thr = prefix[b];                    // full 32-bit threshold key
  const unsigned rem = remain[b];

  float4 v = *reinterpret_cast<const float4*>(in + base);
  float vals[4] = {v.x, v.y, v.z, v.w};
  float res[4];
#pragma unroll
  for (int j = 0; j < 4; ++j) {
    int c = (tid * 4 + j) & (kC - 1);
    unsigned key = orderedKey(vals[j] * bst[c]);
    float r = 0.0f;
    if (key > thr) {
      r = vals[j];
    } else if (key == thr) {
      unsigned old = atomicAdd(const_cast<unsigned*>(&tie[b]), 1u);
      if (old < rem) r = vals[j];
    }
    res[j] = r;
  }
  *reinterpret_cast<float4*>(out + base) = make_float4(res[0], res[1], res[2], res[3]);
}

// ---------------------------------------------------------------------------
extern "C" void kernel_launch(void* const* d_in, const int* in_sizes, int n_in,
                              void* d_out, int out_size, void* d_ws, size_t ws_size,
                              hipStream_t stream) {
  (void)in_sizes; (void)n_in; (void)out_size; (void)ws_size;

  const float* in   = (const float*)d_in[0];   // [64,32,32,512] f32
  const float* duty = (const float*)d_in[1];   // [512] f32
  float*       out  = (float*)d_out;           // [64,32,32,512] f32

  unsigned* hist   = (unsigned*)d_ws;          // B * 2048
  unsigned* prefix = hist + kB * kBins;        // B
  unsigned* remain = prefix + kB;              // B
  unsigned* tie    = remain + kB;              // B
  float*    boost  = (float*)(tie + kB);       // C

  kw_init<<<(kB * kBins + 255) / 256, 256, 0, stream>>>(hist, prefix, remain, tie, boost, duty);

  const int histGrid = kB * kHistBlocksPerSample;    // 2048 blocks
  kw_hist<0><<<histGrid, 256, 0, stream>>>(in, boost, hist, prefix);
  kw_select<<<kB, 256, 0, stream>>>(hist, prefix, remain, kBins, 11);
  kw_hist<1><<<histGrid, 256, 0, stream>>>(in, boost, hist, prefix);
  kw_select<<<kB, 256, 0, stream>>>(hist, prefix, remain, kBins, 11);
  kw_hist<2><<<histGrid, 256, 0, stream>>>(in, boost, hist, prefix);
  kw_select<<<kB, 256, 0, stream>>>(hist, prefix, remain, 1024, 10);

  kw_output<<<kB * (kN / kChunk), 256, 0, stream>>>(in, boost, prefix, remain, tie, out);
}